// CPF_3221225472194
// MI455X (gfx1250) — compile-verified
//
#include <hip/hip_runtime.h>
#include <hip/hip_bf16.h>
#include <math.h>

// ---------------------------------------------------------------------------
// Problem constants (match reference)
// ---------------------------------------------------------------------------
#define NN 100000
#define CC 16
#define GG 2
#define LL 2
#define EE 3200000
#define FF 512
#define HH 64

#define W1_PSTRIDE 160          // words per k-pair row of w1 in LDS (128 + 32 pad)
#define W2_PSTRIDE 32           // words per k-pair row of w2 in LDS
#define HID_STRIDE 66           // words per row of the hidden tile in LDS

typedef float v2f __attribute__((ext_vector_type(2)));
typedef float v8f __attribute__((ext_vector_type(8)));

// ---------------------------------------------------------------------------
// Monotone float <-> uint mapping for atomic float max (handles negatives)
// ---------------------------------------------------------------------------
__device__ __forceinline__ unsigned f2ou(float f) {
    unsigned u = __float_as_uint(f);
    return (u & 0x80000000u) ? ~u : (u | 0x80000000u);
}
__device__ __forceinline__ float ou2f(unsigned u) {
    return __uint_as_float((u & 0x80000000u) ? (u & 0x7FFFFFFFu) : ~u);
}

// ---------------------------------------------------------------------------
// PLP kernels (segment softmax + weighted scatter-sum)
// ---------------------------------------------------------------------------
__global__ void k_init(unsigned* __restrict__ nmax, float* __restrict__ nsum,
                       float* __restrict__ hnext) {
    int i = blockIdx.x * blockDim.x + threadIdx.x;
    if (i < NN) { nmax[i] = 0x007FFFFFu; /* f2ou(-inf) */ nsum[i] = 0.0f; }
    if (i < NN * CC) hnext[i] = 0.0f;
}

__global__ void k_segmax(const float* __restrict__ e, const int* __restrict__ dst,
                         unsigned* __restrict__ nmax) {
    int i = blockIdx.x * blockDim.x + threadIdx.x;
    if (i >= EE) return;
    atomicMax(&nmax[dst[i]], f2ou(e[i]));
}

__global__ void k_segsum(const float* __restrict__ e, const int* __restrict__ dst,
                         const unsigned* __restrict__ nmax, float* __restrict__ nsum) {
    int i = blockIdx.x * blockDim.x + threadIdx.x;
    if (i >= EE) return;
    int d = dst[i];
    atomicAdd(&nsum[d], expf(e[i] - ou2f(nmax[d])));
}

__global__ void k_agg(const float* __restrict__ e, const int* __restrict__ src,
                      const int* __restrict__ dst, const unsigned* __restrict__ nmax,
                      const float* __restrict__ nsum, const float* __restrict__ hcur,
                      float* __restrict__ hnext) {
    int i = blockIdx.x * blockDim.x + threadIdx.x;
    if (i >= EE) return;
    int d = dst[i], s = src[i];
    float a = expf(e[i] - ou2f(nmax[d])) / nsum[d];
    const float4* hs = (const float4*)(hcur + (size_t)s * CC);   // 64B aligned gather
    float4 v0 = hs[0], v1 = hs[1], v2 = hs[2], v3 = hs[3];
    float* o = hnext + (size_t)d * CC;
    atomicAdd(o + 0,  v0.x * a); atomicAdd(o + 1,  v0.y * a);
    atomicAdd(o + 2,  v0.z * a); atomicAdd(o + 3,  v0.w * a);
    atomicAdd(o + 4,  v1.x * a); atomicAdd(o + 5,  v1.y * a);
    atomicAdd(o + 6,  v1.z * a); atomicAdd(o + 7,  v1.w * a);
    atomicAdd(o + 8,  v2.x * a); atomicAdd(o + 9,  v2.y * a);
    atomicAdd(o + 10, v2.z * a); atomicAdd(o + 11, v2.w * a);
    atomicAdd(o + 12, v3.x * a); atomicAdd(o + 13, v3.y * a);
    atomicAdd(o + 14, v3.z * a); atomicAdd(o + 15, v3.w * a);
}

__global__ void k_mask(const float* __restrict__ one_hot, const int* __restrict__ mask,
                       float* __restrict__ h) {
    int i = blockIdx.x * blockDim.x + threadIdx.x;
    if (i >= NN * CC) return;
    int node = i >> 4;
    h[i] = mask[node] ? one_hot[i] : h[i];    // h*(1-m) + one_hot*m, m in {0,1}
}

// ---------------------------------------------------------------------------
// Fused MLP (fp32 WMMA 16x16x4) + attention/alpha combine.
// One wave = two 16-row tiles (M=32). 6250 tiles = 3125 pairs, exact.
// w1/w2 staged into LDS in k-pair-interleaved layout so every B fragment is
// a single aligned ds_load_b64 (no register repacking before v_wmma).
// ---------------------------------------------------------------------------
__global__ __launch_bounds__(256) void k_mlp_combine(
    const float* __restrict__ feat, const float* __restrict__ w1,
    const float* __restrict__ b1,   const float* __restrict__ w2,
    const float* __restrict__ b2,   const float* __restrict__ alpha,
    const float* __restrict__ att,  const float* __restrict__ h0,
    const float* __restrict__ h1,   float* __restrict__ out, int n_pairs)
{
    __shared__ float sW1p[(FF / 2) * W1_PSTRIDE];      // 160 KB: w1, pair-interleaved
    __shared__ float sW2p[(HH / 2) * W2_PSTRIDE];      //   4 KB: w2, pair-interleaved
    __shared__ float sHid[8][2][16 * HID_STRIDE];      //  66 KB: per-wave hidden tiles

    int t = threadIdx.x;

    // ---- stage w1 -> pair-interleaved: sW1p[p*160 + 2n + (k&1)] = w1[k][n], p=k/2
    for (int i = t; i < (FF / 2) * (HH / 4); i += 256) {   // 4096 quad-groups
        int p  = i >> 4;
        int nq = (i & 15) << 2;
        float4 x = *(const float4*)&w1[(size_t)p * 2 * HH + nq];        // k=2p
        float4 y = *(const float4*)&w1[(size_t)p * 2 * HH + HH + nq];   // k=2p+1
        float4 lo = { x.x, y.x, x.y, y.y };
        float4 hi = { x.z, y.z, x.w, y.w };
        float* d = &sW1p[p * W1_PSTRIDE + 2 * nq];
        *(float4*)(d + 0) = lo;
        *(float4*)(d + 4) = hi;
    }
    // ---- stage w2 -> pair-interleaved: sW2p[p*32 + 2n + (k&1)] = w2[k][n]
    for (int i = t; i < (HH / 2) * CC; i += 256) {         // 512 pairs
        int p = i >> 4, n = i & 15;
        v2f v; v.x = w2[(size_t)p * 2 * CC + n];
        v.y = w2[(size_t)p * 2 * CC + CC + n];
        *(v2f*)&sW2p[p * W2_PSTRIDE + 2 * n] = v;
    }
    __syncthreads();

    int wave = t >> 5;
    int lane = t & 31;
    int pair = blockIdx.x * 8 + wave;
    if (pair >= n_pairs) return;            // wave-uniform: EXEC stays all-ones

    int tile0 = pair * 2;
    int tile1 = tile0 + 1;

    int m     = lane & 15;                  // A-frag row / D column index
    int n     = lane & 15;
    int hi16  = lane >> 4;                  // 0 or 1: which half-wave
    int khalf = hi16 << 1;                  // 0 or 2: K pair offset for this half
    int moff  = hi16 << 3;                  // 0 or 8: C/D row offset

    // ---- GEMM1: two [16,512] tiles x [512,64] -> 8 accumulators ----
    v8f acc0 = {}, acc1 = {}, acc2 = {}, acc3 = {};
    v8f acc4 = {}, acc5 = {}, acc6 = {}, acc7 = {};
    const float* arow0 = feat + ((size_t)tile0 * 16 + m) * FF;
    const float* arow1 = feat + ((size_t)tile1 * 16 + m) * FF;
    for (int k = 0; k < FF; k += 4) {
        int kb = k + khalf;
        v2f a0 = *(const v2f*)(arow0 + kb);
        v2f a1 = *(const v2f*)(arow1 + kb);
        const float* bp = &sW1p[((k >> 1) + hi16) * W1_PSTRIDE + 2 * n];
        v2f b0 = *(const v2f*)(bp + 0);     // cols n
        v2f b1v = *(const v2f*)(bp + 32);   // cols n+16
        v2f b2v = *(const v2f*)(bp + 64);   // cols n+32
        v2f b3v = *(const v2f*)(bp + 96);   // cols n+48
        acc0 = __builtin_amdgcn_wmma_f32_16x16x4_f32(false, a0, false, b0,  (short)0, acc0, false, false);
        acc1 = __builtin_amdgcn_wmma_f32_16x16x4_f32(false, a0, false, b1v, (short)0, acc1, false, false);
        acc2 = __builtin_amdgcn_wmma_f32_16x16x4_f32(false, a0, false, b2v, (short)0, acc2, false, false);
        acc3 = __builtin_amdgcn_wmma_f32_16x16x4_f32(false, a0, false, b3v, (short)0, acc3, false, false);
        acc4 = __builtin_amdgcn_wmma_f32_16x16x4_f32(false, a1, false, b0,  (short)0, acc4, false, false);
        acc5 = __builtin_amdgcn_wmma_f32_16x16x4_f32(false, a1, false, b1v, (short)0, acc5, false, false);
        acc6 = __builtin_amdgcn_wmma_f32_16x16x4_f32(false, a1, false, b2v, (short)0, acc6, false, false);
        acc7 = __builtin_amdgcn_wmma_f32_16x16x4_f32(false, a1, false, b3v, (short)0, acc7, false, false);
    }

    // ---- bias + ReLU, spill both hidden tiles to LDS row-major [16][66] ----
    float bb0 = b1[n + 0], bb1 = b1[n + 16], bb2 = b1[n + 32], bb3 = b1[n + 48];
    {
        float* hd0 = &sHid[wave][0][0];
        float* hd1 = &sHid[wave][1][0];
        for (int r = 0; r < 8; ++r) {
            int mm = r + moff;
            hd0[mm * HID_STRIDE + n +  0] = fmaxf(acc0[r] + bb0, 0.0f);
            hd0[mm * HID_STRIDE + n + 16] = fmaxf(acc1[r] + bb1, 0.0f);
            hd0[mm * HID_STRIDE + n + 32] = fmaxf(acc2[r] + bb2, 0.0f);
            hd0[mm * HID_STRIDE + n + 48] = fmaxf(acc3[r] + bb3, 0.0f);
            hd1[mm * HID_STRIDE + n +  0] = fmaxf(acc4[r] + bb0, 0.0f);
            hd1[mm * HID_STRIDE + n + 16] = fmaxf(acc5[r] + bb1, 0.0f);
            hd1[mm * HID_STRIDE + n + 32] = fmaxf(acc6[r] + bb2, 0.0f);
            hd1[mm * HID_STRIDE + n + 48] = fmaxf(acc7[r] + bb3, 0.0f);
        }
    }

    // ---- GEMM2 + epilogue, per tile ----
    float bc = b2[n];
    for (int tt = 0; tt < 2; ++tt) {
        const float* hd = &sHid[wave][tt][0];
        v8f acc = {};
        for (int k = 0; k < HH; k += 4) {
            int kb = k + khalf;
            v2f a = *(const v2f*)(hd + m * HID_STRIDE + kb);          // ds_load_b64
            v2f b = *(const v2f*)&sW2p[((k >> 1) + hi16) * W2_PSTRIDE + 2 * n];
            acc = __builtin_amdgcn_wmma_f32_16x16x4_f32(false, a, false, b, (short)0, acc, false, false);
        }
        int tile = tt ? tile1 : tile0;
        for (int r = 0; r < 8; ++r) {
            int node = tile * 16 + r + moff;
            float mlp = acc[r] + bc;
            float al  = alpha[node];
            float sa  = 1.0f / (1.0f + expf(-al));      // sigmoid(alpha)
            float sna = 1.0f - sa;                      // sigmoid(-alpha)
            float a0 = att[node * 2 + 0], a1 = att[node * 2 + 1];
            float mx = fmaxf(a0, a1);
            float e0 = expf(a0 - mx), e1 = expf(a1 - mx);
            float inv = 1.0f / (e0 + e1);
            float logit = h0[(size_t)node * CC + n] * (e0 * inv)
                        + h1[(size_t)node * CC + n] * (e1 * inv);
            out[(size_t)node * CC + n] = sa * logit + sna * mlp;
        }
    }
}

// ---------------------------------------------------------------------------
// Launcher
// ---------------------------------------------------------------------------
extern "C" void kernel_launch(void* const* d_in, const int* in_sizes, int n_in,
                              void* d_out, int out_size, void* d_ws, size_t ws_size,
                              hipStream_t stream) {
    const float* features      = (const float*)d_in[0];
    const float* label_init    = (const float*)d_in[1];
    const float* labels_onehot = (const float*)d_in[2];
    const float* alpha         = (const float*)d_in[3];
    const float* attention     = (const float*)d_in[4];
    const float* e_edge        = (const float*)d_in[5];
    const float* w1            = (const float*)d_in[6];
    const float* b1            = (const float*)d_in[7];
    const float* w2            = (const float*)d_in[8];
    const float* b2            = (const float*)d_in[9];
    const int*   src           = (const int*)d_in[10];
    const int*   dst           = (const int*)d_in[11];
    const int*   train_mask    = (const int*)d_in[12];
    float* out = (float*)d_out;

    // workspace layout (floats): nmax[N] | nsum[N] | hA[N*C] | hout0[N*C] | hout1[N*C]
    float* ws = (float*)d_ws;
    unsigned* nmax = (unsigned*)ws;
    float* nsum  = ws + NN;
    float* hA    = ws + 2 * NN;
    float* hout0 = hA + NN * CC;
    float* hout1 = hout0 + NN * CC;

    const int TB = 256;
    const int gridNC = (NN * CC + TB - 1) / TB;
    const int gridE  = (EE + TB - 1) / TB;

    for (int g = 0; g < GG; ++g) {
        float* hout = g ? hout1 : hout0;
        const float* hcur = label_init;
        for (int l = 0; l < LL; ++l) {
            float* hnext = (l == LL - 1) ? hout : hA;
            const float* ee = e_edge + ((size_t)l * GG + g) * EE;
            const int* sg = src + (size_t)g * EE;
            const int* dg = dst + (size_t)g * EE;
            k_init  <<<gridNC, TB, 0, stream>>>(nmax, nsum, hnext);
            k_segmax<<<gridE,  TB, 0, stream>>>(ee, dg, nmax);
            k_segsum<<<gridE,  TB, 0, stream>>>(ee, dg, nmax, nsum);
            k_agg   <<<gridE,  TB, 0, stream>>>(ee, sg, dg, nmax, nsum, hcur, hnext);
            k_mask  <<<gridNC, TB, 0, stream>>>(labels_onehot, train_mask, hnext);
            hcur = hnext;
        }
    }

    const int n_pairs = NN / 32;                        // 3125 tile-pairs, exact
    const int blocks  = (n_pairs + 7) / 8;              // 8 waves (pairs) per block
    k_mlp_combine<<<blocks, TB, 0, stream>>>(features, w1, b1, w2, b2,
                                             alpha, attention, hout0, hout1,
                                             out, n_pairs);
}